// StructNConv2D_e_with_d_24429773979965
// MI455X (gfx1250) — compile-verified
//
#include <hip/hip_runtime.h>
#include <hip/hip_bf16.h>

#define EPS 1e-20f

typedef __attribute__((ext_vector_type(2))) float v2f;
typedef __attribute__((ext_vector_type(8))) float v8f;
typedef __attribute__((ext_vector_type(4))) unsigned int v4u;
typedef __attribute__((ext_vector_type(4))) int v4i;
typedef __attribute__((ext_vector_type(8))) int v8i;

static constexpr int B_ = 2, C_ = 4, H_ = 512, W_ = 512;
static constexpr int HP = H_ + 2, WP = W_ + 2;    // padded grid 514x514
static constexpr int NCH = 16;                    // C*4 conv channels
static constexpr int KS = 5, NTAP = 25;
static constexpr int NWTS = NCH * NCH * NTAP;     // 6400 conv weights

// --------------------------------------------------------------------------
// Kernel 1: dual 3x3 "max-pool with indices" on the once-padded grid.
// Replicates jnp.argmax semantics: row-major window order, first strict max
// wins; candidates outside the 514-grid score -inf (gather value 0);
// candidates in the first pad ring score 0 with value 0.
// --------------------------------------------------------------------------
__global__ void pool_kernel(const float* __restrict__ d,
                            const float* __restrict__ cd,
                            float* __restrict__ d_max, float* __restrict__ cd_max,
                            float* __restrict__ d_min, float* __restrict__ cd_min)
{
    int idx = blockIdx.x * blockDim.x + threadIdx.x;
    int total = B_ * C_ * HP * WP;
    if (idx >= total) return;
    int x  = idx % WP;
    int y  = (idx / WP) % HP;
    int bc = idx / (WP * HP);   // b*C + c

    const float NEGINF = -__builtin_inff();
    float bs1 = NEGINF, bs2 = NEGINF;
    float d1 = 0.f, c1 = 0.f, d2 = 0.f, c2 = 0.f;
#pragma unroll
    for (int i = 0; i < 3; ++i) {
#pragma unroll
        for (int j = 0; j < 3; ++j) {
            int py = y + i - 1, px = x + j - 1;
            float dv = 0.f, cv = 0.f, s1, s2;
            if (py < 0 || py >= HP || px < 0 || px >= WP) {
                s1 = NEGINF; s2 = NEGINF;          // second-pad: -inf score
            } else {
                int oy = py - 1, ox = px - 1;       // original coords
                if (oy >= 0 && oy < H_ && ox >= 0 && ox < W_) {
                    dv = d[(bc * H_ + oy) * W_ + ox];
                    cv = cd[(bc * H_ + oy) * W_ + ox];
                }
                s1 = dv * cv;
                s2 = cv / (dv + EPS);
            }
            if (s1 > bs1) { bs1 = s1; d1 = dv; c1 = cv; }
            if (s2 > bs2) { bs2 = s2; d2 = dv; c2 = cv; }
        }
    }
    d_max[idx] = d1; cd_max[idx] = c1;
    d_min[idx] = d2; cd_min[idx] = c2;
}

// --------------------------------------------------------------------------
// Kernel 2: directional ratio / pow / argmax-of-2 / confidence blend.
// Writes p2 = ce2*e2 and ce2 in [B,16,H,W] layout (channel = c*4+k).
// --------------------------------------------------------------------------
__global__ void prop_kernel(const float* __restrict__ e,
                            const float* __restrict__ ce,
                            const float* __restrict__ w_prop,
                            const float* __restrict__ w_efd,
                            const float* __restrict__ d_max, const float* __restrict__ cd_max,
                            const float* __restrict__ d_min, const float* __restrict__ cd_min,
                            float* __restrict__ p2, float* __restrict__ ce2out)
{
    int idx = blockIdx.x * blockDim.x + threadIdx.x;
    int total = B_ * C_ * H_ * W_;
    if (idx >= total) return;
    int w = idx % W_;
    int h = (idx / W_) % H_;
    int c = (idx / (W_ * H_)) % C_;
    int b = idx / (W_ * H_ * C_);

    // near deltas relative to padded center (h+1,w+1); far = negation
    const int dy_n[4] = {-1, -1, -1, 0};
    const int dx_n[4] = {-1,  0,  1, 1};

    int base = (b * C_ + c) * HP * WP;
    int cy = h + 1, cx = w + 1;
    float wp = w_prop[c];

    for (int k = 0; k < 4; ++k) {
        int ni = base + (cy + dy_n[k]) * WP + (cx + dx_n[k]);
        int fi = base + (cy - dy_n[k]) * WP + (cx - dx_n[k]);
        float dmin_n = d_min[ni], dmax_n = d_max[ni];
        float cmin_n = cd_min[ni], cmax_n = cd_max[ni];
        float dmin_f = d_min[fi], dmax_f = d_max[fi];
        float cmin_f = cd_min[fi], cmax_f = cd_max[fi];

        float r0 = dmin_n / (dmax_f + EPS);
        float r1 = dmax_n / (dmin_f + EPS);
        r0 = fminf(fmaxf(r0, EPS), 1.0f);
        r1 = fminf(fmaxf(r1, EPS), 1.0f);
        float w0 = w_efd[(c * 2 + 0) * 4 + k];
        float w1 = w_efd[(c * 2 + 1) * 4 + k];
        float dm0 = powf(r0, w0);
        float dm1 = powf(r1, w1);
        float cm0 = cmin_n * cmax_f;
        float cm1 = cmax_n * cmin_f;
        // jnp.argmax over axis of size 2: index 1 only on strict >
        int j = ((cm1 / dm1) > (cm0 / dm0)) ? 1 : 0;
        float efd  = j ? dm1 : dm0;
        float cefd = j ? cm1 : cm0;

        size_t ei = (((size_t)(b * C_ + c) * 4 + k) * H_ + h) * W_ + w;
        float ev = e[ei], cv = ce[ei];
        float denom = wp * cv + cefd;
        float e2 = (wp * cv * ev + cefd * efd) / (denom + EPS);
        float c2 = denom / (wp + 1.0f);

        size_t oi = ((size_t)(b * NCH + c * 4 + k) * H_ + h) * W_ + w;
        p2[oi]     = c2 * e2;
        ce2out[oi] = c2;
    }
}

// --------------------------------------------------------------------------
// Kernel 3: deterministic reduction of the 6400 conv weights.
// --------------------------------------------------------------------------
__global__ void sumw_kernel(const float* __restrict__ wts, float* __restrict__ out)
{
    __shared__ float red[256];
    float s = 0.f;
    for (int i = threadIdx.x; i < NWTS; i += 256) s += wts[i];
    red[threadIdx.x] = s;
    __syncthreads();
    for (int off = 128; off > 0; off >>= 1) {
        if ((int)threadIdx.x < off) red[threadIdx.x] += red[threadIdx.x + off];
        __syncthreads();
    }
    if (threadIdx.x == 0) out[0] = red[0];
}

// --------------------------------------------------------------------------
// Kernel 4: dual 5x5 conv (16->16 ch) as implicit GEMM on V_WMMA_F32_16X16X4.
//   D[16 co x 16 px] += A[16 co x 4 ci] * B[4 ci x 16 px]  over 25 taps x 4.
// 256 threads = 8 waves; wave ly owns pixel row h0+ly, 16 px wide.
// Fragment layouts per ISA 7.12.2 (f32):
//   A: lane M = lane&15 ; K = vgpr + 2*(lane>>4)
//   B: lane N = lane&15 ; K = vgpr + 2*(lane>>4)
//   C/D: vgpr r -> M = r + 8*(lane>>4), N = lane&15
// A lane's two K slots are adjacent channels (c0, c0+1), so all LDS operand
// arrays are packed as channel-pair float2 -> every fragment is one aligned
// ds_load_b64, conflict-free (lane half q-offset lands 32 banks away).
// Conv weights arrive via the Tensor Data Mover (tensor_load_to_lds, D# per
// ISA ch.8) issued by wave 0, then get repacked LDS->LDS behind a barrier.
// --------------------------------------------------------------------------
__global__ __launch_bounds__(256) void conv_wmma_kernel(
    const float* __restrict__ p2, const float* __restrict__ ce2,
    const float* __restrict__ wts, const float* __restrict__ sumw,
    float* __restrict__ e_out, float* __restrict__ ce_out)
{
    __shared__ float  sWraw[NWTS];          // raw TDM landing zone, 25.6 KB
    __shared__ float2 sW2[NTAP][8][NCH];    // [tap][q=kc*2+hh][co] 25.6 KB
    __shared__ float2 sP2[8][12][20];       // p2 tile+halo, ch-paired 15.36 KB
    __shared__ float2 sC2[8][12][20];       // ce2 tile+halo, ch-paired 15.36 KB

    const int tid = threadIdx.x;
    const int w0 = blockIdx.x * 16;
    const int h0 = blockIdx.y * 8;
    const int b  = blockIdx.z;

    // ---- wave 0: issue TDM load of the 6400-float weight block into LDS ----
    if (tid < 32) {
        unsigned long long ga = (unsigned long long)(const void*)wts;
        v4u g0;
        g0[0] = 1u;                                       // count=1 valid D#
        g0[1] = (unsigned)(size_t)&sWraw[0];              // lds_addr (bytes)
        g0[2] = (unsigned)(ga & 0xFFFFFFFFu);             // global_addr[31:0]
        g0[3] = (unsigned)((ga >> 32) & 0x01FFFFFFu)      // global_addr[56:32]
                | (2u << 30);                             // type = 2 ("image")
        v8i g1;
        g1[0] = 0x2 << 16;                 // workgroup_mask=0, data_size=4B
        g1[1] = (int)((unsigned)NWTS << 16);  // tensor_dim0[15:0] in [31:16]
        g1[2] = 1 << 16;                   // tensor_dim0[31:16]=0, tensor_dim1=1
        g1[3] = (int)((unsigned)NWTS << 16);  // tile_dim0 = 6400
        g1[4] = 1;                         // tile_dim1 = 1, tile_dim2 = 0
        g1[5] = NWTS;                      // tensor_dim0_stride
        g1[6] = (int)((unsigned)NWTS << 16);  // tensor_dim1_stride low bits
        g1[7] = 0;
        v4i z4 = {0, 0, 0, 0};
#if defined(__clang_major__) && (__clang_major__ >= 23)
        v8i z8 = {0, 0, 0, 0, 0, 0, 0, 0};
        __builtin_amdgcn_tensor_load_to_lds(g0, g1, z4, z4, z8, 0);
#else
        __builtin_amdgcn_tensor_load_to_lds(g0, g1, z4, z4, 0);
#endif
        __builtin_amdgcn_s_wait_tensorcnt(0);
    }

    // ---- all waves: stage input tiles (12x20 halo) as channel-pair float2 --
    for (int i = tid; i < 8 * 12 * 20; i += 256) {
        int q = i / 240;            // channel pair (2q, 2q+1)
        int r = i % 240;
        int y = r / 20, x = r % 20;
        int gy = h0 - 2 + y, gx = w0 - 2 + x;
        float2 pv = make_float2(0.f, 0.f);
        float2 cv = make_float2(0.f, 0.f);
        if (gy >= 0 && gy < H_ && gx >= 0 && gx < W_) {
            size_t gi0 = ((size_t)(b * NCH + 2 * q) * H_ + gy) * W_ + gx;
            size_t gi1 = gi0 + (size_t)H_ * W_;     // channel 2q+1
            pv = make_float2(p2[gi0], p2[gi1]);
            cv = make_float2(ce2[gi0], ce2[gi1]);
        }
        sP2[q][y][x] = pv;
        sC2[q][y][x] = cv;
    }
    __syncthreads();   // raw weights (TDM) + input tiles now visible

    // ---- repack raw weights into channel-paired fragment layout -----------
    for (int i = tid; i < NWTS; i += 256) {
        int j   = i & 1;            // K slot within pair
        int co  = (i >> 1) & 15;
        int q   = (i >> 5) & 7;     // kc*2 + lane-half
        int tap = i >> 8;
        int ci  = (q >> 1) * 4 + (q & 1) * 2 + j;
        ((float*)&sW2[tap][q][co])[j] = sWraw[(co * NCH + ci) * NTAP + tap];
    }
    __syncthreads();

    const int lane = tid & 31;
    const int ly   = tid >> 5;      // wave id == pixel row inside tile
    const int n    = lane & 15;     // N (pixel) and M (out-channel row for A)
    const int hh   = lane >> 4;     // lane half

    v8f accN = {0.f, 0.f, 0.f, 0.f, 0.f, 0.f, 0.f, 0.f};
    v8f accD = {0.f, 0.f, 0.f, 0.f, 0.f, 0.f, 0.f, 0.f};

    for (int kh = 0; kh < KS; ++kh) {
        const int ry = ly + kh;
        for (int kw = 0; kw < KS; ++kw) {
            const int tap = kh * KS + kw;
            const int rx  = kw + n;
#pragma unroll
            for (int kc = 0; kc < 4; ++kc) {
                const int q = kc * 2 + hh;    // channel pair (K = 2q, 2q+1)
                float2 aw = sW2[tap][q][n];
                float2 bw = sP2[q][ry][rx];
                float2 cw = sC2[q][ry][rx];
                v2f a, bp, bc;
                a.x = aw.x;  a.y = aw.y;      // A: M=n, K=c0 / c0+1
                bp.x = bw.x; bp.y = bw.y;     // B: N=n, K=c0 / c0+1
                bc.x = cw.x; bc.y = cw.y;
                accN = __builtin_amdgcn_wmma_f32_16x16x4_f32(
                    false, a, false, bp, (short)0, accN, false, false);
                accD = __builtin_amdgcn_wmma_f32_16x16x4_f32(
                    false, a, false, bc, (short)0, accD, false, false);
            }
        }
    }

    const float sw = sumw[0] + EPS;
    const int h = h0 + ly;
#pragma unroll
    for (int r = 0; r < 8; ++r) {
        const int co   = r + 8 * hh;
        const float nm = accN[r];
        const float dn = accD[r];
        size_t oi = ((size_t)(b * NCH + co) * H_ + h) * W_ + (w0 + n);
        e_out[oi]  = nm / (dn + EPS);
        ce_out[oi] = dn / sw;
    }
}

// --------------------------------------------------------------------------
extern "C" void kernel_launch(void* const* d_in, const int* in_sizes, int n_in,
                              void* d_out, int out_size, void* d_ws, size_t ws_size,
                              hipStream_t stream)
{
    (void)in_sizes; (void)n_in; (void)out_size; (void)ws_size;
    const float* d    = (const float*)d_in[0];
    const float* cd   = (const float*)d_in[1];
    const float* e    = (const float*)d_in[2];
    const float* ce   = (const float*)d_in[3];
    const float* wpr  = (const float*)d_in[4];
    const float* wefd = (const float*)d_in[5];
    const float* cw   = (const float*)d_in[6];

    float* ws = (float*)d_ws;
    const size_t np = (size_t)B_ * C_ * HP * WP;    // 2,113,568 per pooled array
    const size_t nx = (size_t)B_ * NCH * H_ * W_;   // 8,388,608 per [B,16,H,W]
    float* d_max  = ws;
    float* cd_max = ws + np;
    float* d_min  = ws + 2 * np;
    float* cd_min = ws + 3 * np;
    float* p2     = ws + 4 * np;
    float* c2     = p2 + nx;
    float* sumw   = c2 + nx;

    float* e_out  = (float*)d_out;
    float* ce_out = e_out + nx;

    sumw_kernel<<<1, 256, 0, stream>>>(cw, sumw);

    {
        int total = (int)np;
        pool_kernel<<<(total + 255) / 256, 256, 0, stream>>>(
            d, cd, d_max, cd_max, d_min, cd_min);
    }
    {
        int total = B_ * C_ * H_ * W_;
        prop_kernel<<<(total + 255) / 256, 256, 0, stream>>>(
            e, ce, wpr, wefd, d_max, cd_max, d_min, cd_min, p2, c2);
    }
    {
        dim3 grid(W_ / 16, H_ / 8, B_);
        conv_wmma_kernel<<<grid, 256, 0, stream>>>(p2, c2, cw, sumw, e_out, ce_out);
    }
}